// FeatureCombination_88201448391101
// MI455X (gfx1250) — compile-verified
//
#include <hip/hip_runtime.h>

typedef __attribute__((ext_vector_type(2))) float v2f;
typedef __attribute__((ext_vector_type(8))) float v8f;
typedef __attribute__((ext_vector_type(4))) float f4;

#define KNN    16
#define NQ     4096
#define NR     8192
#define CFEAT  256
#define QTILE  16
#define RTILE  16
#define NWAVES 8

#define CHUNK_REFS   512                       // refs staged per async chunk
#define NCHUNKS      (NR / CHUNK_REFS)         // 16
#define TILES_PER_CH (CHUNK_REFS / RTILE)      // 32
#define RND_PER_CH   (TILES_PER_CH / NWAVES)   // 4
#define CHUNK_FLOATS (CHUNK_REFS * 3)          // 1536 floats = 6 KB

// ---- async global->LDS path (guarded: falls back to sync staging) ---------
#if defined(__AMDGCN__) && \
    __has_builtin(__builtin_amdgcn_global_load_async_to_lds_b64) && \
    __has_builtin(__builtin_amdgcn_s_wait_asynccnt)
#define HAVE_ASYNC_LDS 1
typedef int v2i __attribute__((ext_vector_type(2)));
typedef __attribute__((address_space(1))) v2i as1_v2i;   // global int2
typedef __attribute__((address_space(3))) v2i as3_v2i;   // LDS int2
#else
#define HAVE_ASYNC_LDS 0
#endif

// Stage one 512-ref chunk of xyz_prev (6144 B) into an LDS buffer.
// Every thread moves exactly 24 bytes = 3 x b64 -> uniform ASYNCcnt of 3.
__device__ __forceinline__
void stage_chunk(const float* __restrict__ xyz_prev, float* lbuf,
                 int chunk, int tid)
{
    const float* g = xyz_prev + (size_t)chunk * CHUNK_FLOATS + tid * 6;
    float*       l = lbuf + tid * 6;
#if HAVE_ASYNC_LDS
    as1_v2i* gp = (as1_v2i*)(void*)g;
    as3_v2i* lp = (as3_v2i*)(void*)l;
    __builtin_amdgcn_global_load_async_to_lds_b64(gp, lp, 0,  0);
    __builtin_amdgcn_global_load_async_to_lds_b64(gp, lp, 8,  0);
    __builtin_amdgcn_global_load_async_to_lds_b64(gp, lp, 16, 0);
#else
    #pragma unroll
    for (int i = 0; i < 6; ++i) l[i] = g[i];
#endif
}

// ---------------------------------------------------------------------------
// Kernel 1: KNN. One workgroup (8 waves / 256 threads) owns one 16-query
// tile. dist'(m,n) = |r_n|^2 - 2 q_m . r_n  (|q_m|^2 dropped: rank-invariant)
// -> one V_WMMA_F32_16X16X4_F32 per 16x16 tile with
//    A[m] = (-2qx, -2qy, -2qz, 1),  B[:,n] = (rx, ry, rz, |r|^2).
// xyz_prev is streamed through double-buffered LDS with async copies.
// ---------------------------------------------------------------------------
__global__ __launch_bounds__(256)
void knn_wmma_kernel(const float* __restrict__ xyz_prev,
                     const float* __restrict__ xyz_cur,
                     int* __restrict__ knn_out)
{
    __shared__ float xyzbuf[2][CHUNK_FLOATS];           // 12 KB (double buffer)
    __shared__ float distLDS[NWAVES][QTILE][RTILE];     //  8 KB
    __shared__ float mdist[QTILE][RTILE][KNN];          // 16 KB
    __shared__ int   midx [QTILE][RTILE][KNN];          // 16 KB

    const int tid  = threadIdx.x;
    const int wave = tid >> 5;
    const int lane = tid & 31;
    const int hi   = lane >> 4;   // K-half for A/B fragments
    const int lo   = lane & 15;   // row (A) / col (B) within tile

    const int qbase = blockIdx.x * QTILE;

    // ---- A fragment (branchless; EXEC stays all-ones into the WMMA) ----
    const float* qp = xyz_cur + (size_t)(qbase + lo) * 3;
    float qx = qp[0], qy = qp[1], qz = qp[2];
    v2f a;
    a.x = hi ? (-2.0f * qz) : (-2.0f * qx);
    a.y = hi ? 1.0f         : (-2.0f * qy);

    // ---- per-thread sorted top-16 (ascending), static indices -> VGPRs ----
    float bd[KNN]; int bi[KNN];
    #pragma unroll
    for (int s = 0; s < KNN; ++s) { bd[s] = 3.0e38f; bi[s] = 0; }

    const int selr = tid >> 4;   // query row this thread filters
    const int selc = tid & 15;   // ref column (mod 16) this thread filters

    stage_chunk(xyz_prev, xyzbuf[0], 0, tid);           // prime the pipeline

    for (int ch = 0; ch < NCHUNKS; ++ch) {
        const int buf = ch & 1;
        if (ch + 1 < NCHUNKS)                            // overlap next copy
            stage_chunk(xyz_prev, xyzbuf[buf ^ 1], ch + 1, tid);
#if HAVE_ASYNC_LDS
        if (ch + 1 < NCHUNKS) __builtin_amdgcn_s_wait_asynccnt(3);
        else                  __builtin_amdgcn_s_wait_asynccnt(0);
#endif
        __syncthreads();                                 // chunk ch visible

        const float* xb = xyzbuf[buf];
        for (int p = 0; p < RND_PER_CH; ++p) {
            // ---- B fragment for this wave's ref tile, from LDS ----
            const int tIn      = p * NWAVES + wave;      // tile within chunk
            const int refLocal = tIn * RTILE + lo;
            float rx = xb[refLocal * 3 + 0];
            float ry = xb[refLocal * 3 + 1];
            float rz = xb[refLocal * 3 + 2];
            float r2 = fmaf(rx, rx, fmaf(ry, ry, rz * rz));
            v2f b;
            b.x = hi ? rz : rx;
            b.y = hi ? r2 : ry;

            v8f c = {};
            c = __builtin_amdgcn_wmma_f32_16x16x4_f32(
                    /*neg_a=*/false, a, /*neg_b=*/false, b,
                    /*c_mod=*/(short)0, c, /*reuse_a=*/false, /*reuse_b=*/false);

            // D layout: lane holds rows v + 8*hi (v=0..7), column = lo
            #pragma unroll
            for (int v = 0; v < 8; ++v)
                distLDS[wave][v + 8 * hi][lo] = c[v];
            __syncthreads();

            // ---- stream the 8 fresh tiles through the register top-16 ----
            const int tileBase = ch * TILES_PER_CH + p * NWAVES;
            #pragma unroll
            for (int w = 0; w < NWAVES; ++w) {
                float d = distLDS[w][selr][selc];
                int   i = (tileBase + w) * RTILE + selc;
                if (d < bd[KNN - 1]) {
                    bd[KNN - 1] = d; bi[KNN - 1] = i;
                    #pragma unroll
                    for (int s = KNN - 1; s > 0; --s) {  // one bubble pass
                        if (bd[s] < bd[s - 1]) {
                            float td = bd[s]; bd[s] = bd[s - 1]; bd[s - 1] = td;
                            int   ti = bi[s]; bi[s] = bi[s - 1]; bi[s - 1] = ti;
                        }
                    }
                }
            }
            __syncthreads();
        }
    }

    // ---- dump per-thread sorted lists to LDS ----
    #pragma unroll
    for (int s = 0; s < KNN; ++s) {
        mdist[selr][selc][s] = bd[s];
        midx [selr][selc][s] = bi[s];
    }
    __syncthreads();

    // ---- 16-way merge: the 16 lanes of a half own one query row ----
    // (rows 2w / 2w+1 live in lanes 0-15 / 16-31 of wave w; xor masks < 16
    //  keep the shuffle reduction inside each half)
    int ptr = 0;
    for (int round = 0; round < KNN; ++round) {
        float md = mdist[selr][selc][ptr];
        int   mi = midx [selr][selc][ptr];
        int   ml = selc;
        #pragma unroll
        for (int off = 1; off < 16; off <<= 1) {
            float od = __shfl_xor(md, off, 32);
            int   oi = __shfl_xor(mi, off, 32);
            int   ol = __shfl_xor(ml, off, 32);
            if (od < md || (od == md && ol < ml)) { md = od; mi = oi; ml = ol; }
        }
        if (selc == 0)
            knn_out[(qbase + selr) * KNN + round] = mi;
        if (ml == selc) ++ptr;   // winner advances its sorted-list head
    }
}

// ---------------------------------------------------------------------------
// Kernel 2: gather + combine. Pure bandwidth: 134 MB of b128 stores, gather
// reads come from L2 (feat_prev is 8 MB << 192 MB L2).
// out[row, 0:256]   = feat_prev[idx[row]] - feat_cur[row>>4]
// out[row, 256:512] = feat_cur[row>>4]
// ---------------------------------------------------------------------------
__global__ __launch_bounds__(256)
void gather_combine_kernel(const float* __restrict__ feat_prev,
                           const float* __restrict__ feat_cur,
                           const int* __restrict__ knn_idx,
                           float* __restrict__ out)
{
    const int tid = threadIdx.x;
    const int row = blockIdx.x * 2 + (tid >> 7);   // 0 .. 65535
    const int l   = tid & 127;                     // float4 slot in 512-wide row
    const int q   = row >> 4;

    const f4* prev4 = (const f4*)feat_prev;
    const f4* cur4  = (const f4*)feat_cur;
    f4* out4 = (f4*)out;

    if (l < 64) {
        const int ridx = knn_idx[row];
        f4 pv = prev4[(size_t)ridx * (CFEAT / 4) + l];
        f4 cv = cur4 [(size_t)q    * (CFEAT / 4) + l];
        out4[(size_t)row * 128 + l] = pv - cv;
    } else {
        f4 cv = cur4[(size_t)q * (CFEAT / 4) + (l - 64)];
        out4[(size_t)row * 128 + l] = cv;
    }
}

// ---------------------------------------------------------------------------
extern "C" void kernel_launch(void* const* d_in, const int* in_sizes, int n_in,
                              void* d_out, int out_size, void* d_ws, size_t ws_size,
                              hipStream_t stream)
{
    const float* xyz_prev  = (const float*)d_in[0];
    const float* xyz_cur   = (const float*)d_in[1];
    const float* feat_prev = (const float*)d_in[2];
    const float* feat_cur  = (const float*)d_in[3];
    float* out = (float*)d_out;

    int* knn_idx = (int*)d_ws;   // 4096*16 ints = 256 KB scratch

    knn_wmma_kernel<<<NQ / QTILE, 256, 0, stream>>>(xyz_prev, xyz_cur, knn_idx);
    gather_combine_kernel<<<(NQ * KNN) / 2, 256, 0, stream>>>(feat_prev, feat_cur,
                                                              knn_idx, out);
}